// SSAaligner_38749194944985
// MI455X (gfx1250) — compile-verified
//
#include <hip/hip_runtime.h>
#include <cstdint>

// Problem constants (reference: N=M=2048, D=64).
#define NROW 2048
#define NCOL 2048
#define DDIM 64
#define TS   64      // 64x64 output tile per block in the distance kernel
#define PADW 68      // LDS row stride in floats: 64 data + 4 pad dwords (TDM pad_enable)

typedef float v2f  __attribute__((ext_vector_type(2)));
typedef float v8f  __attribute__((ext_vector_type(8)));
typedef unsigned int u32x4 __attribute__((ext_vector_type(4)));
typedef int   i32x8 __attribute__((ext_vector_type(8)));
typedef int   i32x4 __attribute__((ext_vector_type(4)));

// ---------------------------------------------------------------------------
// TDM descriptor: load a 64x64 f32 tile (row stride 64 elements in global)
// into LDS with 4-dword padding after every 64 dwords -> LDS stride 68 floats.
// D# layout per CDNA5 ISA.
// This toolchain exposes the 6-arg tensor_load_to_lds builtin (therock-10.0 /
// clang-23 form): (u32x4 g0, i32x8 g1, i32x4 g2, i32x4 g3, i32x8 extra, i32 cpol).
// ---------------------------------------------------------------------------
__device__ __forceinline__ void tdm_load_tile64x64(const float* gsrc, uint32_t lds_byte_off) {
  unsigned long long ga = (unsigned long long)(uintptr_t)gsrc;
  u32x4 g0;
  g0[0] = 1u;                                   // count=1 (valid), user mode
  g0[1] = lds_byte_off;                         // lds_addr
  g0[2] = (unsigned)(ga & 0xFFFFFFFFu);         // global_addr[31:0]
  g0[3] = (unsigned)((ga >> 32) & 0x1FFFFFFu)   // global_addr[56:32]
        | (2u << 30);                           // type=2 (image)
  i32x8 g1;
  // data_size=2 (4B), pad_enable=1, pad_interval=5 (64 dwords), pad_amount=3 (4 dwords)
  g1[0] = (int)((2u << 16) | (1u << 20) | (5u << 22) | (3u << 25));
  g1[1] = (int)((unsigned)DDIM << 16);          // tensor_dim0[15:0] @ bits[63:48]
  g1[2] = (int)((unsigned)NROW << 16);          // tensor_dim1[15:0] @ bits[95:80]
  g1[3] = (int)((unsigned)DDIM << 16);          // tile_dim0 @ bits[127:112]
  g1[4] = TS;                                   // tile_dim1 @ bits[143:128]
  g1[5] = DDIM;                                 // tensor_dim0_stride[31:0]
  g1[6] = 0;                                    // dim0_stride hi / dim1_stride lo
  g1[7] = 0;
  i32x4 z4 = {};                                // 2D tensor: groups 2/3 unused
  i32x8 z8 = {};                                // extra descriptor words: zero
  __builtin_amdgcn_tensor_load_to_lds(g0, g1, z4, z4, z8, 0);
}

// Online softmax-statistics merge: (m,s) <- merge((m,s),(m2,s2))
__device__ __forceinline__ void ms_merge(float& m, float& s, float m2, float s2) {
  float M = fmaxf(m, m2);
  s = s * __expf(m - M) + s2 * __expf(m2 - M);
  m = M;
}

// ---------------------------------------------------------------------------
// Kernel 1: S[i,j] = -sum_d |x[i,d]-y[j,d]|.  Grid (32,32), 256 thr.
// Tiles staged via TDM into padded LDS; each thread computes a 4x4 sub-tile.
// ---------------------------------------------------------------------------
__global__ __launch_bounds__(256)
void ssa_dist_kernel(const float* __restrict__ zx, const float* __restrict__ zy,
                     float* __restrict__ S) {
  __shared__ float lx[TS * PADW];
  __shared__ float ly[TS * PADW];
  const int bx = blockIdx.x, by = blockIdx.y;
  const int t  = threadIdx.x;

  if (t < 32) {  // uniform per-wave branch: wave 0 issues both DMAs
    tdm_load_tile64x64(zx + (size_t)by * TS * DDIM, (uint32_t)(uintptr_t)&lx[0]);
    tdm_load_tile64x64(zy + (size_t)bx * TS * DDIM, (uint32_t)(uintptr_t)&ly[0]);
    __builtin_amdgcn_s_wait_tensorcnt(0);
  }
  __syncthreads();

  const int tx = t & 15, ty = t >> 4;
  float acc[4][4] = {};
  #pragma unroll
  for (int dc = 0; dc < DDIM; dc += 4) {
    float4 xr[4], yc[4];
    #pragma unroll
    for (int r = 0; r < 4; ++r)
      xr[r] = *(const float4*)&lx[(ty * 4 + r) * PADW + dc];   // 272B row stride: 16B aligned
    #pragma unroll
    for (int c = 0; c < 4; ++c)
      yc[c] = *(const float4*)&ly[(tx * 4 + c) * PADW + dc];
    #pragma unroll
    for (int r = 0; r < 4; ++r)
      #pragma unroll
      for (int c = 0; c < 4; ++c)
        acc[r][c] += fabsf(xr[r].x - yc[c].x) + fabsf(xr[r].y - yc[c].y)
                   + fabsf(xr[r].z - yc[c].z) + fabsf(xr[r].w - yc[c].w);
  }

  const int j0 = bx * TS + tx * 4;
  #pragma unroll
  for (int r = 0; r < 4; ++r) {
    const int i = by * TS + ty * 4 + r;
    float4 o = make_float4(-acc[r][0], -acc[r][1], -acc[r][2], -acc[r][3]);
    *(float4*)&S[(size_t)i * NCOL + j0] = o;
  }
}

// ---------------------------------------------------------------------------
// Kernel 2: per-row max & sum(exp(s-max)).  Grid 2048 blocks (one row each).
// ---------------------------------------------------------------------------
__global__ __launch_bounds__(256)
void ssa_row_stats(const float* __restrict__ S, float* __restrict__ rmax,
                   float* __restrict__ rsum) {
  const int i = blockIdx.x, t = threadIdx.x;
  float m = -__builtin_inff(), s = 0.f;
  for (int j = t; j < NCOL; j += 256) {
    float v = S[(size_t)i * NCOL + j];
    float M = fmaxf(m, v);
    s = s * __expf(m - M) + __expf(v - M);
    m = M;
  }
  __shared__ float sm[256], ss[256];
  sm[t] = m; ss[t] = s;
  __syncthreads();
  for (int off = 128; off > 0; off >>= 1) {
    if (t < off) ms_merge(sm[t], ss[t], sm[t + off], ss[t + off]);
    __syncthreads();
  }
  if (t == 0) { rmax[i] = sm[0]; rsum[i] = ss[0]; }
}

// ---------------------------------------------------------------------------
// Kernel 3: per-column partial stats.  Grid (64,8): 32 cols x 256-row chunk.
// Lanes of a wave cover 32 consecutive columns -> coalesced 128B reads.
// ---------------------------------------------------------------------------
__global__ __launch_bounds__(256)
void ssa_col_partial(const float* __restrict__ S, float* __restrict__ cmp,
                     float* __restrict__ csp) {
  const int t  = threadIdx.x;
  const int c  = t & 31;
  const int rl = t >> 5;                    // 0..7 row lanes
  const int j  = blockIdx.x * 32 + c;
  const int i0 = blockIdx.y * 256;
  float m = -__builtin_inff(), s = 0.f;
  for (int k = rl; k < 256; k += 8) {
    float v = S[(size_t)(i0 + k) * NCOL + j];
    float M = fmaxf(m, v);
    s = s * __expf(m - M) + __expf(v - M);
    m = M;
  }
  __shared__ float sm[256], ss[256];
  sm[t] = m; ss[t] = s;
  __syncthreads();
  for (int off = 4; off > 0; off >>= 1) {
    if (rl < off) {
      int o = (rl + off) * 32 + c;
      ms_merge(sm[t], ss[t], sm[o], ss[o]);
    }
    __syncthreads();
  }
  if (rl == 0) {
    cmp[(size_t)blockIdx.y * NCOL + j] = sm[c];
    csp[(size_t)blockIdx.y * NCOL + j] = ss[c];
  }
}

// Kernel 4: merge the 8 row-chunk partials per column.
__global__ __launch_bounds__(256)
void ssa_col_merge(const float* __restrict__ cmp, const float* __restrict__ csp,
                   float* __restrict__ cmax, float* __restrict__ csum) {
  const int j = blockIdx.x * 256 + threadIdx.x;
  float m = cmp[j], s = csp[j];
  #pragma unroll
  for (int q = 1; q < 8; ++q) ms_merge(m, s, cmp[(size_t)q * NCOL + j], csp[(size_t)q * NCOL + j]);
  cmax[j] = m; csum[j] = s;
}

// ---------------------------------------------------------------------------
// Kernel 5: w = a+b-ab; per-block partial sums of w and w*s.
// Block bid handles exactly row bid.  The 256 per-thread partials are reduced
// with V_WMMA_F32_16X16X4_F32 (A = ones 16x4, chained C) — since A is all-ones,
// sum over all D elements == 16 * sum(inputs) independent of VGPR mapping.
// ---------------------------------------------------------------------------
__global__ __launch_bounds__(256)
void ssa_combine(const float* __restrict__ S,
                 const float* __restrict__ rmax, const float* __restrict__ rsum,
                 const float* __restrict__ cmax, const float* __restrict__ csum,
                 float* __restrict__ part_w, float* __restrict__ part_ws) {
  const int t = threadIdx.x;
  const int bid = blockIdx.x;               // == row index i
  const float rm = rmax[bid], rs = rsum[bid];
  float sw = 0.f, sws = 0.f;
  #pragma unroll
  for (int k = 0; k < 8; ++k) {
    const int j = k * 256 + t;
    const float s = S[(size_t)bid * NCOL + j];
    const float a = __expf(s - rm) / rs;
    const float b = __expf(s - cmax[j]) / csum[j];
    const float w = a + b - a * b;
    sw  += w;
    sws += w * s;
  }

  __shared__ float rw[256], rws[256];
  rw[t] = sw; rws[t] = sws;
  __syncthreads();

  if (t < 32) {  // uniform branch: wave 0 only, EXEC all-1s for WMMA
    v2f a1; a1[0] = 1.0f; a1[1] = 1.0f;     // ones A (16x4)
    v8f cw = {}; v8f cs2 = {};
    #pragma unroll
    for (int q = 0; q < 4; ++q) {
      v2f bw, bs;
      bw[0] = rw[q * 64 + t];       bw[1] = rw[q * 64 + 32 + t];
      bs[0] = rws[q * 64 + t];      bs[1] = rws[q * 64 + 32 + t];
      cw  = __builtin_amdgcn_wmma_f32_16x16x4_f32(false, a1, false, bw, (short)0, cw,  false, false);
      cs2 = __builtin_amdgcn_wmma_f32_16x16x4_f32(false, a1, false, bs, (short)0, cs2, false, false);
    }
    float tw = cw[0]+cw[1]+cw[2]+cw[3]+cw[4]+cw[5]+cw[6]+cw[7];
    float ts = cs2[0]+cs2[1]+cs2[2]+cs2[3]+cs2[4]+cs2[5]+cs2[6]+cs2[7];
    #pragma unroll
    for (int off = 16; off > 0; off >>= 1) {
      tw += __shfl_xor(tw, off, 32);
      ts += __shfl_xor(ts, off, 32);
    }
    if (t == 0) {
      part_w[bid]  = tw * (1.0f / 16.0f);   // every input counted in 16 of 256 D slots
      part_ws[bid] = ts * (1.0f / 16.0f);
    }
  }
}

// Kernel 6: deterministic final reduce of 2048 block partials + divide.
__global__ __launch_bounds__(256)
void ssa_finish(const float* __restrict__ part_w, const float* __restrict__ part_ws,
                float* __restrict__ out) {
  const int t = threadIdx.x;
  float w = 0.f, ws = 0.f;
  for (int k = t; k < NROW; k += 256) { w += part_w[k]; ws += part_ws[k]; }
  __shared__ float aw[256], aws[256];
  aw[t] = w; aws[t] = ws;
  __syncthreads();
  for (int off = 128; off > 0; off >>= 1) {
    if (t < off) { aw[t] += aw[t + off]; aws[t] += aws[t + off]; }
    __syncthreads();
  }
  if (t == 0) out[0] = aws[0] / aw[0];
}

// ---------------------------------------------------------------------------
extern "C" void kernel_launch(void* const* d_in, const int* in_sizes, int n_in,
                              void* d_out, int out_size, void* d_ws, size_t ws_size,
                              hipStream_t stream) {
  const float* zx = (const float*)d_in[0];
  const float* zy = (const float*)d_in[1];

  // Workspace layout (floats): S | rmax | rsum | cmp[8*2048] | csp[8*2048]
  //                            | cmax | csum | part_w | part_ws   (~17 MB total)
  float* S       = (float*)d_ws;
  float* rmax    = S + (size_t)NROW * NCOL;
  float* rsum    = rmax + NROW;
  float* cmp     = rsum + NROW;
  float* csp     = cmp + 8 * NCOL;
  float* cmax    = csp + 8 * NCOL;
  float* csum    = cmax + NCOL;
  float* part_w  = csum + NCOL;
  float* part_ws = part_w + NROW;

  ssa_dist_kernel<<<dim3(NCOL / TS, NROW / TS), 256, 0, stream>>>(zx, zy, S);
  ssa_row_stats  <<<NROW, 256, 0, stream>>>(S, rmax, rsum);
  ssa_col_partial<<<dim3(NCOL / 32, NROW / 256), 256, 0, stream>>>(S, cmp, csp);
  ssa_col_merge  <<<NCOL / 256, 256, 0, stream>>>(cmp, csp, cmax, csum);
  ssa_combine    <<<NROW, 256, 0, stream>>>(S, rmax, rsum, cmax, csum, part_w, part_ws);
  ssa_finish     <<<1, 256, 0, stream>>>(part_w, part_ws, (float*)d_out);
}